// GINConv_26723286516149
// MI455X (gfx1250) — compile-verified
//
#include <hip/hip_runtime.h>
#include <cstdint>
#include <cstddef>

typedef float    v2f __attribute__((ext_vector_type(2)));
typedef float    v8f __attribute__((ext_vector_type(8)));
typedef uint32_t v4u __attribute__((ext_vector_type(4)));
typedef uint32_t v8u __attribute__((ext_vector_type(8)));

#define D    128
#define LDSS 132   // padded LDS row stride (floats): TDM pad 4 dwords / 128 dwords

// ---------------- Kernel 0: transpose W1,W2 into workspace --------------
// W?T[n*D + k] = W?[k*D + n]  -> B fragments become one aligned b64 load.
__global__ void gin_transpose_w(const float* __restrict__ W1,
                                const float* __restrict__ W2,
                                float* __restrict__ W1T,
                                float* __restrict__ W2T) {
  int idx = blockIdx.x * blockDim.x + threadIdx.x;   // 0 .. 2*16384-1
  int m   = idx >> 14;                               // which matrix
  int e   = idx & 16383;
  int k   = e >> 7;                                  // row in W
  int n   = e & 127;                                 // col in W
  if (m == 0) W1T[(size_t)n * D + k] = W1[(size_t)k * D + n];
  else        W2T[(size_t)n * D + k] = W2[(size_t)k * D + n];
}

// ---------------- Kernel 1: agg = x  (EPS = 0 self term) ----------------
__global__ void gin_init_agg(const float* __restrict__ x,
                             float* __restrict__ agg, int n4) {
  int i = blockIdx.x * blockDim.x + threadIdx.x;
  if (i < n4) ((float4*)agg)[i] = ((const float4*)x)[i];
}

// ---------------- Kernel 2: agg[dst] += w * x[src] ----------------------
// One wave32 per edge: lane l handles floats [4l, 4l+4) of the 128-wide row.
__global__ void gin_scatter(const float* __restrict__ x,
                            const int* __restrict__ ei,
                            const float* __restrict__ ew,
                            float* __restrict__ agg, int E) {
  int g = blockIdx.x * blockDim.x + threadIdx.x;
  int e = g >> 5;
  if (e >= E) return;
  int lane = g & 31;
  int s = ei[e];        // src row
  int d = ei[E + e];    // dst row
  float w = ew[e];
  float4 v = ((const float4*)(x + (size_t)s * D))[lane];
  float* dst = agg + (size_t)d * D + (size_t)lane * 4;
  __hip_atomic_fetch_add(dst + 0, w * v.x, __ATOMIC_RELAXED, __HIP_MEMORY_SCOPE_AGENT);
  __hip_atomic_fetch_add(dst + 1, w * v.y, __ATOMIC_RELAXED, __HIP_MEMORY_SCOPE_AGENT);
  __hip_atomic_fetch_add(dst + 2, w * v.z, __ATOMIC_RELAXED, __HIP_MEMORY_SCOPE_AGENT);
  __hip_atomic_fetch_add(dst + 3, w * v.w, __ATOMIC_RELAXED, __HIP_MEMORY_SCOPE_AGENT);
}

// ---------------- Kernel 3: fused MLP, TDM A-tile + f32 WMMA ------------
// Block = 256 threads = 8 waves; block handles 16 rows. Wave w owns output
// columns [16w,16w+16). A tile staged by the Tensor Data Mover with HW
// padding giving a 132-float LDS row stride (bank-conflict free).
__global__ __launch_bounds__(256) void gin_mlp(const float* __restrict__ agg,
                                               const float* __restrict__ W1T,
                                               const float* __restrict__ b1,
                                               const float* __restrict__ W2T,
                                               const float* __restrict__ b2,
                                               float* __restrict__ y, int N) {
  __shared__ float lds_a[16 * LDSS];
  __shared__ float lds_h[16 * LDSS];

  const int row0 = blockIdx.x * 16;
  const int tid  = threadIdx.x;
  const int wave = tid >> 5;
  const int lane = tid & 31;
  const int half = lane >> 4;   // 0: lanes 0-15, 1: lanes 16-31
  const int lm   = lane & 15;

  // ---- TDM: DMA the 16x128 f32 A-tile (rows of `agg`) into padded LDS ----
  if (wave == 0) {
    uint32_t rows = (uint32_t)(N - row0);
    if (rows > 16u) rows = 16u;
    uint64_t ga = (uint64_t)(uintptr_t)(agg + (size_t)row0 * D);
    uint32_t la = (uint32_t)(uintptr_t)&lds_a[0];

    // D# group 0: count=1 | lds_addr | global_addr[56:0] | type=2
    v4u g0;
    g0.x = 1u;                                   // count=1, user mode, no gather
    g0.y = la;                                   // LDS byte address
    g0.z = (uint32_t)ga;                         // global_addr[31:0]
    g0.w = ((uint32_t)(ga >> 32) & 0x01FFFFFFu)  // global_addr[56:32]
           | (2u << 30);                         // type = 2 ("image")

    // D# group 1: data_size=4B, pad every 128 dwords by 4 dwords (stride 132)
    v8u g1;
    g1.s0 = (2u << 16)            // data_size = 2 -> 4 bytes
          | (1u << 20)            // pad_enable
          | (6u << 22)            // pad_interval code 6 -> 128 dwords
          | (3u << 25);           // pad_amount  code 3 -> 4 dwords
    g1.s1 = (128u << 16);         // tensor_dim0[15:0] = 128 (atomic_barrier=0)
    g1.s2 = (rows & 0xFFFFu) << 16;          // tensor_dim0[31:16]=0 | tensor_dim1 lo
    g1.s3 = ((rows >> 16) & 0xFFFFu)         // tensor_dim1 hi
          | (128u << 16);                    // tile_dim0 = 128
    g1.s4 = 16u;                  // tile_dim1 = 16, tile_dim2 = 0
    g1.s5 = 128u;                 // tensor_dim0_stride[31:0] = 128 elements
    g1.s6 = 0u;                   // stride hi | tensor_dim1_stride lo (unused)
    g1.s7 = 0u;                   // tensor_dim1_stride hi (unused)

    asm volatile("tensor_load_to_lds %0, %1" :: "s"(g0), "s"(g1) : "memory");
    __builtin_amdgcn_s_wait_tensorcnt(0);
  }
  __syncthreads();

  const int ncol = wave * 16 + lm;   // this lane's output column

  // ---- GEMM 1: h = relu(A @ W1 + b1), via V_WMMA_F32_16X16X4_F32 ----
  v8f acc = {};
#pragma unroll
  for (int kb = 0; kb < 32; ++kb) {
    int k = kb * 4 + 2 * half;       // ISA A/B fragment layout for 16x16x4
    v2f a, b;
    a.x = lds_a[lm * LDSS + k];
    a.y = lds_a[lm * LDSS + k + 1];
    float2 bv = *(const float2*)&W1T[(size_t)ncol * D + k];
    b.x = bv.x; b.y = bv.y;
    acc = __builtin_amdgcn_wmma_f32_16x16x4_f32(false, a, false, b,
                                                (short)0, acc, false, false);
  }
  {
    float bias = b1[ncol];
#pragma unroll
    for (int v = 0; v < 8; ++v) {
      int r = v + 8 * half;          // C/D layout: VGPR v -> row v + 8*half
      float h = acc[v] + bias;
      lds_h[r * LDSS + ncol] = h > 0.f ? h : 0.f;
    }
  }
  __syncthreads();

  // ---- GEMM 2: y = h @ W2 + b2 ----
  v8f acc2 = {};
#pragma unroll
  for (int kb = 0; kb < 32; ++kb) {
    int k = kb * 4 + 2 * half;
    v2f a, b;
    a.x = lds_h[lm * LDSS + k];
    a.y = lds_h[lm * LDSS + k + 1];
    float2 bv = *(const float2*)&W2T[(size_t)ncol * D + k];
    b.x = bv.x; b.y = bv.y;
    acc2 = __builtin_amdgcn_wmma_f32_16x16x4_f32(false, a, false, b,
                                                 (short)0, acc2, false, false);
  }
  {
    float bias = b2[ncol];
#pragma unroll
    for (int v = 0; v < 8; ++v) {
      int r = v + 8 * half;
      if (row0 + r < N)
        y[(size_t)(row0 + r) * D + ncol] = acc2[v] + bias;
    }
  }
}

// ------------------------------------------------------------------------
extern "C" void kernel_launch(void* const* d_in, const int* in_sizes, int n_in,
                              void* d_out, int out_size, void* d_ws, size_t ws_size,
                              hipStream_t stream) {
  const float* x  = (const float*)d_in[0];
  const int*   ei = (const int*)d_in[1];   // edge_index [2, E]
  const float* ew = (const float*)d_in[2];
  const float* W1 = (const float*)d_in[3];
  const float* b1 = (const float*)d_in[4];
  const float* W2 = (const float*)d_in[5];
  const float* b2 = (const float*)d_in[6];
  float* y = (float*)d_out;

  const int N = in_sizes[0] / D;
  const int E = in_sizes[2];

  float* agg = (float*)d_ws;               // N*D floats of scratch
  float* W1T = agg + (size_t)N * D;        // 16384 floats
  float* W2T = W1T + (size_t)D * D;        // 16384 floats

  // 0) transpose weights (B-operand friendly layout)
  gin_transpose_w<<<(2 * D * D) / 256, 256, 0, stream>>>(W1, W2, W1T, W2T);

  // 1) agg = x
  int n4 = N * (D / 4);
  gin_init_agg<<<(n4 + 255) / 256, 256, 0, stream>>>(x, agg, n4);

  // 2) scatter-add messages
  long long sthreads = (long long)E * 32;
  gin_scatter<<<(unsigned)((sthreads + 255) / 256), 256, 0, stream>>>(x, ei, ew, agg, E);

  // 3) fused 2-layer MLP: TDM A-tile staging + f32 WMMA
  gin_mlp<<<(N + 15) / 16, 256, 0, stream>>>(agg, W1T, b1, W2T, b2, y, N);
}